// GAT_90744069030460
// MI455X (gfx1250) — compile-verified
//
#include <hip/hip_runtime.h>
#include <hip/hip_bf16.h>

typedef __attribute__((ext_vector_type(2))) float v2f;
typedef __attribute__((ext_vector_type(8))) float v8f;

#define NEG_SLOPE 0.2f

// ---------- helpers ----------

__device__ inline unsigned fkey(float f) {
    // monotonic float -> uint mapping (for atomicMax on signed floats)
    unsigned u = __float_as_uint(f);
    return (u >> 31) ? ~u : (u | 0x80000000u);
}
__device__ inline float funkey(unsigned k) {
    return __uint_as_float((k >> 31) ? (k & 0x7FFFFFFFu) : ~k);
}
__device__ inline void edge_sd(const int* __restrict__ ei, int E, int id, int& s, int& d) {
    if (id < E) { s = ei[id]; d = ei[E + id]; }
    else        { s = d = id - E; }  // appended self-loops
}
__device__ inline float lrelu(float e) { return e >= 0.f ? e : NEG_SLOPE * e; }

// ---------- WMMA fp32 GEMM: H[M,NF] = A[M,K] @ B[K,NF] ----------
// One wave computes a 16 x NF strip. Exact fp32 via v_wmma_f32_16x16x4_f32.

template <int K, int NF>
__global__ __launch_bounds__(256) void gemm_wmma(const float* __restrict__ A,
                                                 const float* __restrict__ B,
                                                 float* __restrict__ H, int M) {
    static_assert(NF % 16 == 0 && K % 4 == 0, "dims");
    constexpr int NT = NF / 16;
    const int wave = (blockIdx.x * blockDim.x + threadIdx.x) >> 5;
    const int lane = threadIdx.x & 31;
    const int tilesM = (M + 15) >> 4;
    if (wave >= tilesM) return;           // wave-uniform: EXEC stays all-ones for WMMA

    const int row0 = wave << 4;
    const int half = lane >> 4;           // 0: K=k0..k0+1 | 1: K=k0+2..k0+3
    const int r    = lane & 15;

    v8f acc[NT];
#pragma unroll
    for (int t = 0; t < NT; ++t) acc[t] = (v8f){0.f,0.f,0.f,0.f,0.f,0.f,0.f,0.f};

    int arow = row0 + r; if (arow >= M) arow = M - 1;   // safety clamp (M%16==0 in practice)
    const float* Aptr = A + (size_t)arow * K;

    for (int k0 = 0; k0 < K; k0 += 4) {
        const int ks = k0 + half * 2;
        v2f a;
        a.x = Aptr[ks];
        a.y = Aptr[ks + 1];
#pragma unroll
        for (int t = 0; t < NT; ++t) {
            const int col = t * 16 + r;
            v2f b;
            b.x = B[(size_t)ks * NF + col];
            b.y = B[(size_t)(ks + 1) * NF + col];
            acc[t] = __builtin_amdgcn_wmma_f32_16x16x4_f32(
                false, a, false, b, (short)0, acc[t], false, false);
        }
    }

    // C/D layout: VGPR j -> row j (lanes 0-15) / row j+8 (lanes 16-31), col = lane&15
    if (row0 + 16 <= M) {
        // full tile: wave-uniform, no per-element exec masking -> straight-line stores
        float* Hbase = H + (size_t)(row0 + half * 8) * NF + r;
#pragma unroll
        for (int t = 0; t < NT; ++t)
#pragma unroll
            for (int j = 0; j < 8; ++j)
                Hbase[(size_t)j * NF + t * 16] = acc[t][j];
    } else {
        // ragged last tile (unused when M % 16 == 0)
#pragma unroll
        for (int t = 0; t < NT; ++t)
#pragma unroll
            for (int j = 0; j < 8; ++j) {
                const int rr = row0 + j + half * 8;
                const int cc = t * 16 + r;
                if (rr < M) H[(size_t)rr * NF + cc] = acc[t][j];
            }
    }
}

// ---------- per-node attention dots: a_src[n]=h[n]·as, a_dst[n]=h[n]·ad ----------

__global__ __launch_bounds__(256) void att_dots(const float* __restrict__ h,
                                                const float* __restrict__ as_,
                                                const float* __restrict__ ad_,
                                                float* __restrict__ a_s,
                                                float* __restrict__ a_d, int N, int F) {
    const int wave = (blockIdx.x * blockDim.x + threadIdx.x) >> 5;
    const int lane = threadIdx.x & 31;
    if (wave >= N) return;
    float s = 0.f, d = 0.f;
    for (int f = lane; f < F; f += 32) {
        const float hv = h[(size_t)wave * F + f];
        s += hv * as_[f];
        d += hv * ad_[f];
    }
#pragma unroll
    for (int off = 16; off > 0; off >>= 1) {
        s += __shfl_xor(s, off, 32);
        d += __shfl_xor(d, off, 32);
    }
    if (lane == 0) { a_s[wave] = s; a_d[wave] = d; }
}

// ---------- zeroing ----------

__global__ void zero_f32(float* __restrict__ p, size_t n) {
    size_t i = (size_t)blockIdx.x * blockDim.x + threadIdx.x;
    if (i < n) p[i] = 0.f;
}
__global__ void zero_u32(unsigned* __restrict__ p, size_t n) {
    size_t i = (size_t)blockIdx.x * blockDim.x + threadIdx.x;
    if (i < n) p[i] = 0u;
}

// ---------- edge passes ----------

__global__ __launch_bounds__(256) void edge_max_k(const int* __restrict__ ei, int E, int ET,
                                                  const float* __restrict__ a_s,
                                                  const float* __restrict__ a_d,
                                                  unsigned* __restrict__ m_u) {
    const int id = blockIdx.x * blockDim.x + threadIdx.x;
    if (id >= ET) return;
    int s, d; edge_sd(ei, E, id, s, d);
    const float e = lrelu(a_s[s] + a_d[d]);
    atomicMax(&m_u[d], fkey(e));
}

__global__ __launch_bounds__(256) void edge_exp_k(const int* __restrict__ ei, int E, int ET,
                                                  const float* __restrict__ a_s,
                                                  const float* __restrict__ a_d,
                                                  const unsigned* __restrict__ m_u,
                                                  float* __restrict__ denom,
                                                  float* __restrict__ pbuf) {
    const int id = blockIdx.x * blockDim.x + threadIdx.x;
    if (id >= ET) return;
    int s, d; edge_sd(ei, E, id, s, d);
    const float e = lrelu(a_s[s] + a_d[d]);
    const float m = funkey(m_u[d]);
    const float p = __expf(e - m);
    pbuf[id] = p;
    atomicAdd(&denom[d], p);
}

// wave-per-edge: agg[dst,:] += alpha * h[src,:]
template <int F>
__global__ __launch_bounds__(256) void edge_aggregate(const int* __restrict__ ei, int E, int ET,
                                                      const float* __restrict__ h,
                                                      const float* __restrict__ pbuf,
                                                      const float* __restrict__ denom,
                                                      float* __restrict__ agg) {
    const long long gid = (long long)blockIdx.x * blockDim.x + threadIdx.x;
    const int edge = (int)(gid >> 5);
    const int lane = (int)(gid & 31);
    if (edge >= ET) return;
    int s, d; edge_sd(ei, E, edge, s, d);
    const float alpha = pbuf[edge] / denom[d];
#pragma unroll
    for (int f = lane; f < F; f += 32)
        atomicAdd(&agg[(size_t)d * F + f], alpha * h[(size_t)s * F + f]);
}

// ---------- epilogue: dst = agg + bias (+ReLU) ----------

template <int F, bool RELU>
__global__ __launch_bounds__(256) void epilogue(const float* __restrict__ agg,
                                                const float* __restrict__ bias,
                                                float* __restrict__ dst, int N) {
    const size_t i = (size_t)blockIdx.x * blockDim.x + threadIdx.x;
    if (i >= (size_t)N * F) return;
    float v = agg[i] + bias[i & (F - 1)];
    if (RELU) v = fmaxf(v, 0.f);
    dst[i] = v;
}

// ---------- host-side layer orchestration ----------

template <int K, int F, bool RELU>
static void gat_layer(const float* in, const float* W, const float* as_, const float* ad_,
                      const float* bias, float* dstOut,
                      float* h, float* agg, float* a_s, float* a_d,
                      unsigned* m_u, float* denom, float* pbuf,
                      const int* ei, int N, int E, hipStream_t stream) {
    const int ET = E + N;
    const int tilesM = (N + 15) / 16;

    gemm_wmma<K, F><<<(tilesM + 7) / 8, 256, 0, stream>>>(in, W, h, N);

    const unsigned wblocks = (unsigned)(((size_t)N * 32 + 255) / 256);
    att_dots<<<wblocks, 256, 0, stream>>>(h, as_, ad_, a_s, a_d, N, F);

    zero_u32<<<(unsigned)((N + 255) / 256), 256, 0, stream>>>(m_u, (size_t)N);
    zero_f32<<<(unsigned)((N + 255) / 256), 256, 0, stream>>>(denom, (size_t)N);
    zero_f32<<<(unsigned)(((size_t)N * F + 255) / 256), 256, 0, stream>>>(agg, (size_t)N * F);

    const unsigned eblocks = (unsigned)((ET + 255) / 256);
    edge_max_k<<<eblocks, 256, 0, stream>>>(ei, E, ET, a_s, a_d, m_u);
    edge_exp_k<<<eblocks, 256, 0, stream>>>(ei, E, ET, a_s, a_d, m_u, denom, pbuf);

    const unsigned ablocks = (unsigned)(((long long)ET * 32 + 255) / 256);
    edge_aggregate<F><<<ablocks, 256, 0, stream>>>(ei, E, ET, h, pbuf, denom, agg);

    epilogue<F, RELU><<<(unsigned)(((size_t)N * F + 255) / 256), 256, 0, stream>>>(agg, bias, dstOut, N);
}

extern "C" void kernel_launch(void* const* d_in, const int* in_sizes, int n_in,
                              void* d_out, int out_size, void* d_ws, size_t ws_size,
                              hipStream_t stream) {
    // inputs: x, edge_index, W1, as1, ad1, b1, W2, as2, ad2, b2, W3, as3, ad3, b3
    const float* x   = (const float*)d_in[0];
    const int*   ei  = (const int*)d_in[1];
    const float* W1  = (const float*)d_in[2];
    const float* as1 = (const float*)d_in[3];
    const float* ad1 = (const float*)d_in[4];
    const float* b1  = (const float*)d_in[5];
    const float* W2  = (const float*)d_in[6];
    const float* as2 = (const float*)d_in[7];
    const float* ad2 = (const float*)d_in[8];
    const float* b2  = (const float*)d_in[9];
    const float* W3  = (const float*)d_in[10];
    const float* as3 = (const float*)d_in[11];
    const float* ad3 = (const float*)d_in[12];
    const float* b3  = (const float*)d_in[13];

    const int N = in_sizes[0] / 128;   // F_in = 128
    const int E = in_sizes[1] / 2;     // edge_index is [2, E]
    const int ET = E + N;

    // output layout: out [N,32] | h1 [N,64] | h2 [N,64]
    float* out3 = (float*)d_out;
    float* h1   = out3 + (size_t)N * 32;
    float* h2   = h1 + (size_t)N * 64;

    // workspace layout
    float*    ws    = (float*)d_ws;
    float*    h     = ws;                       // N*64
    float*    agg   = h + (size_t)N * 64;       // N*64
    float*    a_s   = agg + (size_t)N * 64;     // N
    float*    a_d   = a_s + N;                  // N
    unsigned* m_u   = (unsigned*)(a_d + N);     // N
    float*    denom = (float*)(m_u + N);        // N
    float*    pbuf  = denom + N;                // ET
    (void)ws_size; (void)n_in; (void)out_size; (void)ET;

    // layer 1: [N,128] -> [N,64], ReLU, result -> h1 (in d_out)
    gat_layer<128, 64, true>(x, W1, as1, ad1, b1, h1,
                             h, agg, a_s, a_d, m_u, denom, pbuf, ei, N, E, stream);
    // layer 2: [N,64] -> [N,64], ReLU, result -> h2 (in d_out)
    gat_layer<64, 64, true>(h1, W2, as2, ad2, b2, h2,
                            h, agg, a_s, a_d, m_u, denom, pbuf, ei, N, E, stream);
    // layer 3: [N,64] -> [N,32], no ReLU, result -> out (in d_out)
    gat_layer<64, 32, false>(h2, W3, as3, ad3, b3, out3,
                             h, agg, a_s, a_d, m_u, denom, pbuf, ei, N, E, stream);
}